// ScatterSelfAttention_13408887898530
// MI455X (gfx1250) — compile-verified
//
#include <hip/hip_runtime.h>

// ScatterSelfAttention for MI455X (gfx1250, wave32).
// Sizes fixed by the reference problem.
#define N_NODES 50000
#define N_EDGES 800000
#define DIM     128
#define NH      8
#define DK      16

typedef __attribute__((ext_vector_type(2))) float v2f;
typedef __attribute__((ext_vector_type(8))) float v8f;

// ---------------------------------------------------------------------------
// Dense GEMM: Y[M x 128] = X[M x 128] @ W[128 x 128] + bias, fp32 WMMA.
// One wave -> one 16x16 output tile; 8 waves/block cover all 8 col tiles
// of a 16-row slab. K-loop = 32 x V_WMMA_F32_16X16X4_F32.
// Operand layouts per CDNA5 ISA 7.12.2 (f32 16x16x4):
//   A: lane l holds row (l&15), VGPR{0,1} = K = k0+2*(l>>4), +1
//   B: lane l holds col (l&15), VGPR{0,1} = same K pair
//   C/D: VGPR j, lane l -> row j+8*(l>>4), col (l&15)
// ---------------------------------------------------------------------------
__global__ __launch_bounds__(256) void gemm128_wmma(
    const float* __restrict__ X, const float* __restrict__ W,
    const float* __restrict__ bias, float* __restrict__ Y)
{
  const int lane = threadIdx.x & 31;
  const int wave = threadIdx.x >> 5;          // 0..7 -> column tile
  const int rb   = blockIdx.x * 16;           // row tile base
  const int cb   = wave * 16;                 // col tile base
  const int m    = lane & 15;
  const int hi   = lane >> 4;                 // 0 or 1 (K-pair select)

  const float* xrow = X + (size_t)(rb + m) * DIM;
  v8f acc = {};
#pragma unroll
  for (int k0 = 0; k0 < DIM; k0 += 4) {
    const int kk = k0 + 2 * hi;
    v2f a, b;
    a.x = xrow[kk];
    a.y = xrow[kk + 1];
    b.x = W[(size_t)kk * DIM + cb + m];
    b.y = W[(size_t)(kk + 1) * DIM + cb + m];
    acc = __builtin_amdgcn_wmma_f32_16x16x4_f32(
        /*neg_a=*/false, a, /*neg_b=*/false, b,
        /*c_mod=*/(short)0, acc, /*reuse_a=*/false, /*reuse_b=*/false);
  }
  const float bv = bias[cb + m];
#pragma unroll
  for (int j = 0; j < 8; ++j) {
    Y[(size_t)(rb + j + 8 * hi) * DIM + cb + m] = acc[j] + bv;
  }
}

// Order-preserving float <-> uint encoding so atomicMax(u32) == float max.
__device__ __forceinline__ unsigned fenc(float f) {
  unsigned u = __float_as_uint(f);
  return (u & 0x80000000u) ? ~u : (u | 0x80000000u);
}
__device__ __forceinline__ float fdec(unsigned k) {
  return (k & 0x80000000u) ? __uint_as_float(k & 0x7fffffffu)
                           : __uint_as_float(~k);
}

// Pass 1 over edges: logits (written directly into d_out tail) + seg max.
__global__ __launch_bounds__(256) void edge_logits_kernel(
    const int* __restrict__ ei, const float* __restrict__ att_bias,
    const float* __restrict__ Q, const float* __restrict__ K,
    float* __restrict__ logits, unsigned* __restrict__ segmax)
{
  int tid = blockIdx.x * blockDim.x + threadIdx.x;
  if (tid >= N_EDGES * NH) return;
  const int e  = tid >> 3;
  const int hh = tid & 7;
  const int d  = ei[e];                 // dst (query side, segment id)
  const int s  = ei[N_EDGES + e];       // src (key side)
  const float4* qp = (const float4*)(Q + (size_t)d * DIM + hh * DK);
  const float4* kp = (const float4*)(K + (size_t)s * DIM + hh * DK);
  float dot = 0.f;
#pragma unroll
  for (int i = 0; i < 4; ++i) {
    float4 q4 = qp[i], k4 = kp[i];
    dot += q4.x * k4.x + q4.y * k4.y + q4.z * k4.z + q4.w * k4.w;
  }
  const float lg = 0.25f * dot + att_bias[tid];   // 1/sqrt(16) = 0.25
  logits[tid] = lg;
  atomicMax(&segmax[(size_t)d * NH + hh], fenc(lg));
}

// Pass 2: exp(logit - segmax) accumulated into seg sum.
__global__ __launch_bounds__(256) void edge_expsum_kernel(
    const int* __restrict__ ei, const float* __restrict__ logits,
    const unsigned* __restrict__ segmax, float* __restrict__ segsum)
{
  int tid = blockIdx.x * blockDim.x + threadIdx.x;
  if (tid >= N_EDGES * NH) return;
  const int e  = tid >> 3;
  const int hh = tid & 7;
  const int d  = ei[e];
  const float mx = fdec(segmax[(size_t)d * NH + hh]);
  atomicAdd(&segsum[(size_t)d * NH + hh], __expf(logits[tid] - mx));
}

// Pass 3: score = softmax weight; scatter score * v into agg[N,H,DK].
__global__ __launch_bounds__(256) void edge_scatter_kernel(
    const int* __restrict__ ei, const float* __restrict__ logits,
    const unsigned* __restrict__ segmax, const float* __restrict__ segsum,
    const float* __restrict__ V, float* __restrict__ agg)
{
  int tid = blockIdx.x * blockDim.x + threadIdx.x;
  if (tid >= N_EDGES * NH) return;
  const int e  = tid >> 3;
  const int hh = tid & 7;
  const int d  = ei[e];
  const int s  = ei[N_EDGES + e];
  const float mx = fdec(segmax[(size_t)d * NH + hh]);
  const float sc = __expf(logits[tid] - mx) / segsum[(size_t)d * NH + hh];
  const float4* vp = (const float4*)(V + (size_t)s * DIM + hh * DK);
  float* ap = agg + (size_t)d * DIM + hh * DK;
#pragma unroll
  for (int i = 0; i < 4; ++i) {
    float4 v4 = vp[i];
    atomicAdd(ap + 4 * i + 0, sc * v4.x);
    atomicAdd(ap + 4 * i + 1, sc * v4.y);
    atomicAdd(ap + 4 * i + 2, sc * v4.z);
    atomicAdd(ap + 4 * i + 3, sc * v4.w);
  }
}

extern "C" void kernel_launch(void* const* d_in, const int* in_sizes, int n_in,
                              void* d_out, int out_size, void* d_ws, size_t ws_size,
                              hipStream_t stream) {
  (void)in_sizes; (void)n_in; (void)out_size; (void)ws_size;
  const float* x   = (const float*)d_in[0];
  const int*   ei  = (const int*)d_in[1];     // edge_index [2, E]
  const float* att = (const float*)d_in[2];   // [E, H]
  const float* Wq  = (const float*)d_in[3];
  const float* bq  = (const float*)d_in[4];
  const float* Wk  = (const float*)d_in[5];
  const float* bk  = (const float*)d_in[6];
  const float* Wv  = (const float*)d_in[7];
  const float* bv  = (const float*)d_in[8];
  const float* Wo  = (const float*)d_in[9];
  const float* bo  = (const float*)d_in[10];

  float* out    = (float*)d_out;                       // [N, D]
  float* logits = out + (size_t)N_NODES * DIM;         // [E, H] tail of d_out

  // Workspace layout (floats): Q | K | V | agg | segmax(u32) | segsum
  float*    ws     = (float*)d_ws;
  float*    Q      = ws;                               //  6.4M
  float*    Kp     = ws + (size_t)N_NODES * DIM;       //  6.4M
  float*    Vp     = ws + (size_t)2 * N_NODES * DIM;   //  6.4M
  float*    agg    = ws + (size_t)3 * N_NODES * DIM;   //  6.4M
  unsigned* segmax = (unsigned*)(ws + (size_t)4 * N_NODES * DIM);  // 0.4M
  float*    segsum = ws + (size_t)4 * N_NODES * DIM + (size_t)N_NODES * NH;

  // Zero agg + segmax(0 == minimum of the encoded-float order) + segsum.
  hipMemsetAsync(agg, 0,
                 ((size_t)N_NODES * DIM + 2 * (size_t)N_NODES * NH) * sizeof(float),
                 stream);

  const dim3 blk(256);
  const dim3 gemm_grid(N_NODES / 16);   // 50000 % 16 == 0
  gemm128_wmma<<<gemm_grid, blk, 0, stream>>>(x, Wq, bq, Q);
  gemm128_wmma<<<gemm_grid, blk, 0, stream>>>(x, Wk, bk, Kp);
  gemm128_wmma<<<gemm_grid, blk, 0, stream>>>(x, Wv, bv, Vp);

  const int nthreads = N_EDGES * NH;
  const int nblocks  = (nthreads + 255) / 256;
  edge_logits_kernel<<<nblocks, blk, 0, stream>>>(ei, att, Q, Kp, logits, segmax);
  edge_expsum_kernel<<<nblocks, blk, 0, stream>>>(ei, logits, segmax, segsum);
  edge_scatter_kernel<<<nblocks, blk, 0, stream>>>(ei, logits, segmax, segsum, Vp, agg);

  gemm128_wmma<<<gemm_grid, blk, 0, stream>>>(agg, Wo, bo, out);
}